// FaultTolerantReLU_15109694947575
// MI455X (gfx1250) — compile-verified
//
#include <hip/hip_runtime.h>

// FaultTolerantReLU (noise-free) for MI455X / gfx1250.
//
// Stage-2 of the reference is provably the identity (out_ph is bitwise a row
// of out_table; zero circular distance <=> equal y since y in [0,6], lambda>=19),
// so we only run stage-1: nearest codeword of theta = mod1(x/lambda) over the
// 241-row in_table, then y = relu(x_values[k]).
//
// Nearest-codeword via the chord embedding:
//   argmin_k sum_m circdist(theta_m, t_km)^2
//   ~= argmax_k sum_m cos(2*pi*theta_m)cos(2*pi*t_km) + sin(..)sin(..)
// i.e. argmax of a [N,16] x [16,256] f32 GEMM (K=14 real dims + bias pad),
// executed with V_WMMA_F32_16X16X4_F32 (f32 precision required: score
// separations are ~4e-4 absolute on magnitude ~7; f16/bf16 would misdecode).
//
// Round-2 changes (scheduling only):
//  * A-matrix K-chunks batched into a v2f[4] before the WMMA chain so the
//    4 ds_load_2addr_b32 issue back-to-back instead of load->wait_dscnt(0)->wmma
//    per chunk.
//  * Candidate-tile loop fully unrolled so tile t+1 LDS loads schedule under
//    tile t's WMMA + argmax VALU.

typedef __attribute__((ext_vector_type(2))) float v2f;
typedef __attribute__((ext_vector_type(8))) float v8f;

#define FT_TWO_PI 6.28318530717958647692f
#define FT_SPAD   17      // padded LDS row stride (floats) -> conflict-free A loads
#define FT_SCAND  241     // real codebook rows
#define FT_SCANDP 256     // padded to 16 tiles of 16
#define FT_MDIM   7       // moduli

__global__ __launch_bounds__(256)
void ftrelu_wmma_kernel(const float* __restrict__ x,
                        const float* __restrict__ lambdas,
                        const float* __restrict__ xvals,
                        float* __restrict__ out,
                        int N)
{
    __shared__ float T[FT_SCANDP * FT_SPAD];   // codebook embedding, A-matrix source

    const int tid = threadIdx.x;

    // ---- build codebook embedding table: one candidate row per thread ----
    {
        float inv[FT_MDIM];
#pragma unroll
        for (int m = 0; m < FT_MDIM; ++m) inv[m] = 1.0f / lambdas[m];

        if (tid < FT_SCAND) {
            float xv = xvals[tid];
#pragma unroll
            for (int m = 0; m < FT_MDIM; ++m) {
                float p = xv * inv[m];
                p -= floorf(p);                     // mod1, matches reference
                float a = FT_TWO_PI * p;
                T[tid * FT_SPAD + 2*m]     = __cosf(a);
                T[tid * FT_SPAD + 2*m + 1] = __sinf(a);
            }
            T[tid * FT_SPAD + 14] = 0.0f;           // bias dim (real row)
            T[tid * FT_SPAD + 15] = 0.0f;
        } else if (tid < FT_SCANDP) {
#pragma unroll
            for (int d = 0; d < 16; ++d) T[tid * FT_SPAD + d] = 0.0f;
            T[tid * FT_SPAD + 14] = -1.0e30f;       // poison pad rows via bias
        }
    }
    __syncthreads();

    const int lane = tid & 31;
    const int wv   = tid >> 5;
    const int n16  = lane & 15;   // sample column within tile / A row M
    const int h    = lane >> 4;   // half-wave: selects K pair + D row half

    // Each lane only ever touches moduli m = 2j + h (K-dims 4j+2h, 4j+2h+1).
    float invl[4];
#pragma unroll
    for (int j = 0; j < 4; ++j) {
        int m = 2*j + h;
        invl[j] = (m < FT_MDIM) ? (1.0f / lambdas[m]) : 0.0f;
    }

    const int wavesTotal = (int)((gridDim.x * blockDim.x) >> 5);
    const int gw         = (int)((blockIdx.x * blockDim.x) >> 5) + wv;

    for (int base = gw * 32; base < N; base += wavesTotal * 32) {
        int i0 = base + n16;        if (i0 >= N) i0 = N - 1;
        int i1 = base + 16 + n16;   if (i1 >= N) i1 = N - 1;
        float x0 = x[i0];
        float x1 = x[i1];

        // ---- per-lane sample embeddings for the two B tiles ----
        float e0[8], e1[8];
#pragma unroll
        for (int j = 0; j < 4; ++j) {
            int m = 2*j + h;
            if (m < FT_MDIM) {
                float p0 = x0 * invl[j]; p0 -= floorf(p0); float a0 = FT_TWO_PI * p0;
                e0[2*j]     = __cosf(a0);
                e0[2*j + 1] = __sinf(a0);
                float p1 = x1 * invl[j]; p1 -= floorf(p1); float a1 = FT_TWO_PI * p1;
                e1[2*j]     = __cosf(a1);
                e1[2*j + 1] = __sinf(a1);
            } else {                 // K-dims 14,15: bias dim = 1, pad = 0
                e0[2*j] = 1.0f; e0[2*j + 1] = 0.0f;
                e1[2*j] = 1.0f; e1[2*j + 1] = 0.0f;
            }
        }

        v2f B0[4], B1[4];
#pragma unroll
        for (int c = 0; c < 4; ++c) {
            v2f b0; b0.x = e0[2*c]; b0.y = e0[2*c + 1]; B0[c] = b0;
            v2f b1; b1.x = e1[2*c]; b1.y = e1[2*c + 1]; B1[c] = b1;
        }

        float bv0 = -3.0e38f, bv1 = -3.0e38f;
        int   bi0 = 0,        bi1 = 0;

        // ---- sweep 16 candidate tiles; D = sum_c A_c x B_c via chained WMMA ----
#pragma unroll
        for (int t = 0; t < 16; ++t) {
            const float* row = &T[(t * 16 + n16) * FT_SPAD];

            // batch all 4 K-chunk A loads so ds_loads issue back-to-back
            v2f a4[4];
#pragma unroll
            for (int c = 0; c < 4; ++c) {
                a4[c].x = row[4*c + 2*h];
                a4[c].y = row[4*c + 2*h + 1];
            }

            v8f acc0 = {};
            v8f acc1 = {};
#pragma unroll
            for (int c = 0; c < 4; ++c) {
                acc0 = __builtin_amdgcn_wmma_f32_16x16x4_f32(
                           false, a4[c], false, B0[c], (short)0, acc0, false, false);
                acc1 = __builtin_amdgcn_wmma_f32_16x16x4_f32(
                           false, a4[c], false, B1[c], (short)0, acc1, false, false);
            }

            // lane holds candidates 16t + 8h + r (r = 0..7), column = its sample
#pragma unroll
            for (int r = 0; r < 8; ++r) {
                int cand = t * 16 + 8*h + r;
                float s0 = acc0[r];
                if (s0 > bv0) { bv0 = s0; bi0 = cand; }
                float s1 = acc1[r];
                if (s1 > bv1) { bv1 = s1; bi1 = cand; }
            }
        }

        // ---- merge the two half-wave candidate partitions ----
        float ov0 = __shfl_xor(bv0, 16, 32);
        int   oi0 = __shfl_xor(bi0, 16, 32);
        if (ov0 > bv0) { bv0 = ov0; bi0 = oi0; }
        float ov1 = __shfl_xor(bv1, 16, 32);
        int   oi1 = __shfl_xor(bi1, 16, 32);
        if (ov1 > bv1) { bv1 = ov1; bi1 = oi1; }

        int idx = (h == 0) ? bi0 : bi1;
        if (idx > FT_SCAND - 1) idx = FT_SCAND - 1;
        float y = fmaxf(0.0f, xvals[idx]);          // relu(x_values[k]) == output
        int oidx = (h == 0) ? (base + n16) : (base + 16 + n16);
        if (oidx < N) out[oidx] = y;
    }
}

extern "C" void kernel_launch(void* const* d_in, const int* in_sizes, int n_in,
                              void* d_out, int out_size, void* d_ws, size_t ws_size,
                              hipStream_t stream) {
    const float* x       = (const float*)d_in[0];
    const float* lambdas = (const float*)d_in[1];
    const float* xvals   = (const float*)d_in[2];
    float*       out     = (float*)d_out;
    const int    N       = in_sizes[0];

    const int samplesPerBlock = 8 * 32;  // 8 waves x 32 samples per iteration
    int blocks = (N + samplesPerBlock - 1) / samplesPerBlock;
    if (blocks < 1) blocks = 1;
    if (blocks > 4096) blocks = 4096;    // grid-stride loop covers the rest

    ftrelu_wmma_kernel<<<blocks, 256, 0, stream>>>(x, lambdas, xvals, out, N);
}